// LiMNet_83605833384549
// MI455X (gfx1250) — compile-verified
//
#include <hip/hip_runtime.h>
#include <math.h>

typedef __attribute__((ext_vector_type(2))) float v2f;
typedef __attribute__((ext_vector_type(8))) float v8f;

#define B_   128
#define NU_  10000
#define E_   64
#define FU_  8
#define FI_  8
#define IN_  144      // 2E + FU + FI
#define G3E_ 192      // 3*E
#define XSTR 146      // padded LDS row stride in floats (18 mod 64 -> conflict-free groups)

// ------------------------------------------------------------------
// GEMM: gi[g] = X_g (128x144) @ W_ih_g^T (144x192), f32 WMMA 16x16x4
// grid = 16 blocks (g in {0,1} x mt in 0..7), 128 threads = 4 waves
// ------------------------------------------------------------------
__global__ __launch_bounds__(128)
void limnet_gemm(const float* __restrict__ user_mem,
                 const float* __restrict__ item_mem,
                 const int*   __restrict__ user_ids,
                 const int*   __restrict__ item_ids,
                 const float* __restrict__ user_feat,
                 const float* __restrict__ item_feat,
                 const float* __restrict__ u_w_ih,
                 const float* __restrict__ i_w_ih,
                 float* __restrict__ gi_ws)          // [2][128][192]
{
    __shared__ float Xs[16 * XSTR];

    const int g   = blockIdx.x >> 3;   // 0 = user GRU, 1 = item GRU
    const int mt  = blockIdx.x & 7;    // 16-row batch tile
    const int tid = threadIdx.x;

    // Build 16 rows of X_g in LDS.
    // X_u[m] = [user_emb(64) | user_feat(8) | item_emb(64) | item_feat(8)]
    // X_i is the 72-column rotation of X_u.
    for (int idx = tid; idx < 16 * IN_; idx += 128) {
        const int r  = idx / IN_;
        const int c  = idx % IN_;
        const int m  = mt * 16 + r;                 // batch row
        const int cc = (c + (g ? 72 : 0)) % IN_;    // column in X_u space
        float v;
        if (cc < 64) {
            v = user_mem[((size_t)m * NU_ + (size_t)user_ids[m]) * E_ + cc];
        } else if (cc < 72) {
            v = user_feat[m * FU_ + (cc - 64)];
        } else if (cc < 136) {
            v = item_mem[((size_t)m * NU_ + (size_t)item_ids[m]) * E_ + (cc - 72)];
        } else {
            v = item_feat[m * FI_ + (cc - 136)];
        }
        Xs[r * XSTR + c] = v;
    }
    __syncthreads();

    const int lane = tid & 31;
    const int wid  = tid >> 5;
    const int half = lane >> 4;        // 0: lanes 0-15, 1: lanes 16-31
    const int l16  = lane & 15;

    const float* W = (g == 0) ? u_w_ih : i_w_ih;   // [192][144]

    // A fragment source: Xs[l16][k0 + 2*half + {0,1}]
    const float* arow = &Xs[l16 * XSTR + 2 * half];

    for (int t = 0; t < 3; ++t) {                  // 12 N-tiles / 4 waves
        const int nt = wid + 4 * t;                // 0..11
        const float* wrow = &W[(size_t)(nt * 16 + l16) * IN_ + 2 * half];
        v8f acc = {0.f, 0.f, 0.f, 0.f, 0.f, 0.f, 0.f, 0.f};
        #pragma unroll
        for (int kk = 0; kk < 36; ++kk) {          // K = 144 = 36 * 4
            const int k0 = kk * 4;
            v2f a, b;
            a.x = arow[k0];  a.y = arow[k0 + 1];
            b.x = wrow[k0];  b.y = wrow[k0 + 1];
            acc = __builtin_amdgcn_wmma_f32_16x16x4_f32(
                false, a, false, b, (short)0, acc, false, false);
        }
        // C/D layout: VGPR r -> (M = 8*half + r, N = l16)
        float* out = gi_ws + (size_t)g * (B_ * G3E_)
                   + (size_t)(mt * 16 + 8 * half) * G3E_ + nt * 16 + l16;
        #pragma unroll
        for (int r = 0; r < 8; ++r) out[(size_t)r * G3E_] = acc[r];
    }
}

// ------------------------------------------------------------------
// GRU nonlinearity + L2 normalize + scatter. h0 == 0 so gh == b_hh.
// grid = 256 blocks (g x batch row), 32 threads = 1 wave per row.
// ------------------------------------------------------------------
__global__ __launch_bounds__(32)
void limnet_combine(const float* __restrict__ gi_ws,
                    const int*   __restrict__ user_ids,
                    const int*   __restrict__ item_ids,
                    const float* __restrict__ u_b_ih, const float* __restrict__ u_b_hh,
                    const float* __restrict__ i_b_ih, const float* __restrict__ i_b_hh,
                    float* __restrict__ out)
{
    const int bx   = blockIdx.x;       // 0..255
    const int g    = bx >> 7;
    const int b    = bx & 127;
    const int lane = threadIdx.x;      // 0..31

    const float* gi   = gi_ws + (size_t)g * (B_ * G3E_) + (size_t)b * G3E_;
    const float* b_ih = g ? i_b_ih : u_b_ih;
    const float* b_hh = g ? i_b_hh : u_b_hh;

    float h[2];
    #pragma unroll
    for (int t = 0; t < 2; ++t) {
        const int j = lane + 32 * t;
        const float ir = gi[j]       + b_ih[j];
        const float iz = gi[j + 64]  + b_ih[j + 64];
        const float in = gi[j + 128] + b_ih[j + 128];
        const float r  = 1.f / (1.f + expf(-(ir + b_hh[j])));
        const float z  = 1.f / (1.f + expf(-(iz + b_hh[j + 64])));
        const float n  = tanhf(in + r * b_hh[j + 128]);
        h[t] = (1.f - z) * n;          // + z*h0 == 0
    }
    float s = h[0] * h[0] + h[1] * h[1];
    #pragma unroll
    for (int off = 16; off > 0; off >>= 1) s += __shfl_xor(s, off, 32);
    const float inv = 1.f / fmaxf(sqrtf(s), 1e-12f);

    const int id = g ? item_ids[b] : user_ids[b];
    float* head = out + (size_t)g * (B_ * E_) + (size_t)b * E_;
    float* mem  = out + 2 * (size_t)(B_ * E_)
                + (size_t)g * ((size_t)B_ * NU_ * E_)
                + ((size_t)b * NU_ + (size_t)id) * E_;
    #pragma unroll
    for (int t = 0; t < 2; ++t) {
        const int j = lane + 32 * t;
        const float v = h[t] * inv;
        head[j] = v;
        mem[j]  = v;
    }
}

extern "C" void kernel_launch(void* const* d_in, const int* in_sizes, int n_in,
                              void* d_out, int out_size, void* d_ws, size_t ws_size,
                              hipStream_t stream) {
    (void)in_sizes; (void)n_in; (void)out_size; (void)ws_size;
    const float* user_mem  = (const float*)d_in[0];
    const float* item_mem  = (const float*)d_in[1];
    const int*   user_ids  = (const int*)  d_in[2];
    const int*   item_ids  = (const int*)  d_in[3];
    const float* user_feat = (const float*)d_in[4];
    const float* item_feat = (const float*)d_in[5];
    const float* u_w_ih    = (const float*)d_in[6];
    /* d_in[7] = u_w_hh: unused, h0 == 0 */
    const float* u_b_ih    = (const float*)d_in[8];
    const float* u_b_hh    = (const float*)d_in[9];
    const float* i_w_ih    = (const float*)d_in[10];
    /* d_in[11] = i_w_hh: unused */
    const float* i_b_ih    = (const float*)d_in[12];
    const float* i_b_hh    = (const float*)d_in[13];

    float* out = (float*)d_out;
    float* gi  = (float*)d_ws;                     // needs 2*128*192*4 = 192 KB

    const size_t memElems = (size_t)B_ * NU_ * E_; // 81,920,000 floats each

    // 1) tiny WMMA GEMMs (independent of the bulk copies)
    hipLaunchKernelGGL(limnet_gemm, dim3(16), dim3(128), 0, stream,
                       user_mem, item_mem, user_ids, item_ids,
                       user_feat, item_feat, u_w_ih, i_w_ih, gi);

    // 2) bandwidth-bound bulk copies of the two memory tensors (~1.31 GB traffic)
    hipMemcpyAsync(out + 2 * (size_t)B_ * E_, user_mem,
                   memElems * sizeof(float), hipMemcpyDeviceToDevice, stream);
    hipMemcpyAsync(out + 2 * (size_t)B_ * E_ + memElems, item_mem,
                   memElems * sizeof(float), hipMemcpyDeviceToDevice, stream);

    // 3) nonlinearity + L2 normalize + scatter updated rows (ordered after copies)
    hipLaunchKernelGGL(limnet_combine, dim3(256), dim3(32), 0, stream,
                       gi, user_ids, item_ids, u_b_ih, u_b_hh, i_b_ih, i_b_hh, out);
}